// HeatDiffusion_27401891349054
// MI455X (gfx1250) — compile-verified
//
#include <hip/hip_runtime.h>
#include <hip/hip_bf16.h>

// ---------------------------------------------------------------------------
// HeatDiffusion on MI455X (gfx1250): expm(-5*adj) via scaling-and-squaring
// with fp32 WMMA (V_WMMA_F32_16X16X4_F32) GEMMs. GEMM tiles are staged with
// CDNA5 async global->LDS copies (GLOBAL_LOAD_ASYNC_TO_LDS_B128, ASYNCcnt)
// into double-buffered LDS. Then top-16 mask + symmetrize + self-loops +
// row-normalize.
// ---------------------------------------------------------------------------

typedef __attribute__((ext_vector_type(2))) float v2f;
typedef __attribute__((ext_vector_type(8))) float v8f;

static constexpr int    NMAT = 4096;
static constexpr float  TVAL = 5.0f;
static constexpr int    TOPK = 16;
static constexpr int    SQUARINGS = 3;          // scale by 2^3
static constexpr int    TAYLOR_MAX = 10;        // terms X^2 .. X^10
static constexpr size_t MATN = (size_t)NMAT * NMAT;

// ---------------------------- init: X, P, Tc --------------------------------
__global__ __launch_bounds__(256) void k_init_expm(
    const float* __restrict__ adj, float* __restrict__ X,
    float* __restrict__ P, float* __restrict__ Tc, float c)
{
    size_t i = (size_t)blockIdx.x * blockDim.x + threadIdx.x;
    if (i >= MATN) return;
    int row = (int)(i >> 12);
    int col = (int)(i & (NMAT - 1));
    float x = c * adj[i];
    X[i]  = x;
    Tc[i] = x;
    P[i]  = x + ((row == col) ? 1.0f : 0.0f);   // I + X
}

// ------------------------------ WMMA GEMM -----------------------------------
// D = alpha * (A @ B);  if ACCUM: P += alpha * (A @ B)
// Block tile 128x128, BK=16, 256 threads (8 waves), wave tile 32x64 (2x4 WMMA)
// LDS double-buffered, filled by per-lane async B128 copies.
static constexpr int BM = 128, BN = 128, BK = 16;
static constexpr int ASTR = BK + 4;   // 20 floats = 80B = 5 x 16B chunks
static constexpr int BSTR = BN + 8;   // 136 floats = 544B = 34 x 16B chunks

__device__ __forceinline__ void async_b128(const float* g, const float* l)
{
    // VDST = LDS byte address (low 32 bits of the generic LDS pointer),
    // VADDR = 64-bit global address, GV mode.
    unsigned lds = (unsigned)(size_t)l;
    unsigned long long ga = (unsigned long long)(size_t)g;
    asm volatile("global_load_async_to_lds_b128 %0, %1, off"
                 :: "v"(lds), "v"(ga) : "memory");
}

__device__ __forceinline__ void wait_async0()
{
    asm volatile("s_wait_asynccnt 0x0" ::: "memory");
}

template <bool ACCUM>
__global__ __launch_bounds__(256) void k_gemm_wmma(
    const float* __restrict__ A, const float* __restrict__ B,
    float* __restrict__ D, float* __restrict__ P, float alpha)
{
    // A tile m-major (rows of BK), B tile k-major (rows of BN); both padded
    // with 16B-multiple strides so async 16B chunks stay contiguous.
    __shared__ __attribute__((aligned(16))) float As[2][BM][ASTR];
    __shared__ __attribute__((aligned(16))) float Bs[2][BK][BSTR];

    const int tid   = threadIdx.x;
    const int lane  = tid & 31;
    const int wid   = tid >> 5;        // 0..7
    const int waveM = wid & 3;         // 4 row-groups of 32
    const int waveN = wid >> 2;        // 2 col-groups of 64
    const int half  = lane >> 4;       // 0/1
    const int l16   = lane & 15;

    const int rowBase = blockIdx.y * BM;
    const int colBase = blockIdx.x * BN;

    v8f acc[2][4];
    #pragma unroll
    for (int mi = 0; mi < 2; ++mi)
        #pragma unroll
        for (int ni = 0; ni < 4; ++ni)
            acc[mi][ni] = (v8f)(0.0f);

    // per-thread chunk mapping: A tile = 512 x 16B chunks, B tile = 512 chunks
    const int aR0 = tid >> 2;            // chunk row for pass 0 (0..63)
    const int aC4 = (tid & 3) * 4;       // float col 0,4,8,12
    const int bK0 = tid >> 5;            // k row for pass 0 (0..7)
    const int bN4 = (tid & 31) * 4;      // float col 0..124

    auto issue_stage = [&](int buf, int kt) {
        #pragma unroll
        for (int p = 0; p < 2; ++p) {
            int r = aR0 + p * 64;
            async_b128(A + (size_t)(rowBase + r) * NMAT + kt + aC4,
                       &As[buf][r][aC4]);
        }
        #pragma unroll
        for (int p = 0; p < 2; ++p) {
            int k = bK0 + p * 8;
            async_b128(B + (size_t)(kt + k) * NMAT + colBase + bN4,
                       &Bs[buf][k][bN4]);
        }
    };

    issue_stage(0, 0);
    int buf = 0;

    for (int kt = 0; kt < NMAT; kt += BK) {
        wait_async0();          // my copies for `buf` have landed in LDS
        __syncthreads();        // everyone's copies landed; prev reads done
        if (kt + BK < NMAT) issue_stage(buf ^ 1, kt + BK);

        const float (*Asb)[ASTR] = As[buf];
        const float (*Bsb)[BSTR] = Bs[buf];

        #pragma unroll
        for (int k4 = 0; k4 < BK / 4; ++k4) {
            // ISA 7.12.2: A 16x4 frag -> lanes 0-15 hold K=0,1; lanes 16-31 K=2,3
            const int kk = k4 * 4 + 2 * half;
            v2f a[2], b[4];
            #pragma unroll
            for (int mi = 0; mi < 2; ++mi) {
                int m = waveM * 32 + mi * 16 + l16;
                a[mi].x = Asb[m][kk];        // contiguous pair -> ds_load_b64
                a[mi].y = Asb[m][kk + 1];
            }
            #pragma unroll
            for (int ni = 0; ni < 4; ++ni) {
                int nn = waveN * 64 + ni * 16 + l16;
                b[ni].x = Bsb[kk][nn];
                b[ni].y = Bsb[kk + 1][nn];
            }
            #pragma unroll
            for (int mi = 0; mi < 2; ++mi)
                #pragma unroll
                for (int ni = 0; ni < 4; ++ni)
                    acc[mi][ni] = __builtin_amdgcn_wmma_f32_16x16x4_f32(
                        false, a[mi], false, b[ni], (short)0, acc[mi][ni],
                        false, false);
        }
        buf ^= 1;
    }

    // epilogue: C/D layout -> VGPR r, lanes 0-15: M=r, lanes 16-31: M=r+8
    #pragma unroll
    for (int mi = 0; mi < 2; ++mi) {
        #pragma unroll
        for (int ni = 0; ni < 4; ++ni) {
            const int col = colBase + waveN * 64 + ni * 16 + l16;
            #pragma unroll
            for (int r = 0; r < 8; ++r) {
                const int row = rowBase + waveM * 32 + mi * 16 + half * 8 + r;
                const float v = alpha * acc[mi][ni][r];
                D[(size_t)row * NMAT + col] = v;
                if (ACCUM) P[(size_t)row * NMAT + col] += v;
            }
        }
    }
}

// --------------------------- top-16 per row ---------------------------------
__global__ __launch_bounds__(256) void k_topk16(
    const float* __restrict__ heat, int* __restrict__ idxOut)
{
    __shared__ float rowv[NMAT];     // 16 KB row cache
    __shared__ float redV[256];
    __shared__ int   redI[256];

    const int row = blockIdx.x;
    const int tid = threadIdx.x;

    #pragma unroll
    for (int c = 0; c < NMAT / 256; ++c)
        rowv[c * 256 + tid] = heat[(size_t)row * NMAT + c * 256 + tid];
    __syncthreads();

    for (int it = 0; it < TOPK; ++it) {
        float bv = -__builtin_inff();
        int   bi = 0;
        #pragma unroll
        for (int c = 0; c < NMAT / 256; ++c) {
            int   j = c * 256 + tid;
            float v = rowv[j];
            if (v > bv || (v == bv && j < bi)) { bv = v; bi = j; }
        }
        redV[tid] = bv; redI[tid] = bi;
        __syncthreads();
        for (int s = 128; s > 0; s >>= 1) {
            if (tid < s) {
                float ov = redV[tid + s]; int oi = redI[tid + s];
                if (ov > redV[tid] || (ov == redV[tid] && oi < redI[tid])) {
                    redV[tid] = ov; redI[tid] = oi;
                }
            }
            __syncthreads();
        }
        if (tid == 0) {
            int b = redI[0];
            idxOut[row * TOPK + it] = b;
            rowv[b] = -__builtin_inff();   // exclude from later rounds
        }
        __syncthreads();
    }
}

// ------------------------- mask bitset build --------------------------------
__global__ __launch_bounds__(256) void k_zero_bits(unsigned* __restrict__ bits)
{
    size_t i = (size_t)blockIdx.x * blockDim.x + threadIdx.x;
    if (i < (size_t)NMAT * (NMAT / 32)) bits[i] = 0u;
}

__global__ __launch_bounds__(256) void k_scatter_bits(
    const int* __restrict__ idx, unsigned* __restrict__ bits)
{
    int g = blockIdx.x * blockDim.x + threadIdx.x;
    if (g >= NMAT * TOPK) return;
    int i = g / TOPK;
    int j = idx[g];
    atomicOr(&bits[(size_t)i * (NMAT / 32) + (j >> 5)], 1u << (j & 31));
}

// ---------------- finalize: mask, symmetrize, +I, normalize -----------------
__global__ __launch_bounds__(256) void k_finalize(
    const float* __restrict__ heat, const unsigned* __restrict__ bits,
    float* __restrict__ out)
{
    __shared__ float red[256];
    const int i   = blockIdx.x;
    const int tid = threadIdx.x;

    float vals[NMAT / 256];
    float psum = 0.0f;
    #pragma unroll
    for (int c = 0; c < NMAT / 256; ++c) {
        const int j = c * 256 + tid;
        unsigned m = (bits[(size_t)i * (NMAT / 32) + (j >> 5)] >> (j & 31)) & 1u;
        m |= (bits[(size_t)j * (NMAT / 32) + (i >> 5)] >> (i & 31)) & 1u;
        float v = 0.0f;
        if (m) v = 0.5f * (heat[(size_t)i * NMAT + j] + heat[(size_t)j * NMAT + i]);
        if (j == i) v += 1.0f;             // self-loop
        vals[c] = v;
        psum += v;
    }
    red[tid] = psum;
    __syncthreads();
    for (int s = 128; s > 0; s >>= 1) {
        if (tid < s) red[tid] += red[tid + s];
        __syncthreads();
    }
    const float inv = 1.0f / red[0];
    #pragma unroll
    for (int c = 0; c < NMAT / 256; ++c)
        out[(size_t)i * NMAT + c * 256 + tid] = vals[c] * inv;
}

// ---------------------------------------------------------------------------
extern "C" void kernel_launch(void* const* d_in, const int* in_sizes, int n_in,
                              void* d_out, int out_size, void* d_ws, size_t ws_size,
                              hipStream_t stream) {
    (void)in_sizes; (void)n_in; (void)out_size; (void)ws_size;
    const float* adj = (const float*)d_in[0];
    float* out = (float*)d_out;

    char* base = (char*)d_ws;
    const size_t MB = MATN * sizeof(float);           // 64 MB per matrix
    float*    W0   = (float*)(base);                  // X, later squaring scratch
    float*    W1   = (float*)(base + MB);             // P (exp accumulator)
    float*    W2   = (float*)(base + 2 * MB);         // Taylor term
    int*      idx  = (int*)(base + 3 * MB);           // 4096*16 ints
    unsigned* bits = (unsigned*)(base + 3 * MB + (size_t)NMAT * TOPK * sizeof(int));

    const dim3 blk(256);
    const dim3 gridE((unsigned)((MATN + 255) / 256));
    const dim3 gridG(NMAT / BN, NMAT / BM);

    // X = (-T / 2^s) * adj ; P = I + X ; Tc = X
    const float cscale = -TVAL / (float)(1 << SQUARINGS);
    k_init_expm<<<gridE, blk, 0, stream>>>(adj, W0, W1, W2, cscale);

    // Taylor: Tk = Tk-1 @ X / k, P += Tk   (k = 2 .. TAYLOR_MAX)
    float* X  = W0;
    float* P  = W1;
    float* Tc = W2;
    float* Tn = out;                                  // d_out doubles as scratch
    for (int k = 2; k <= TAYLOR_MAX; ++k) {
        k_gemm_wmma<true><<<gridG, blk, 0, stream>>>(Tc, X, Tn, P, 1.0f / (float)k);
        float* t = Tc; Tc = Tn; Tn = t;
    }

    // Squaring: P <- P @ P, three times (X buffer is dead -> scratch).
    float* S = W0;
    for (int s = 0; s < SQUARINGS; ++s) {
        k_gemm_wmma<false><<<gridG, blk, 0, stream>>>(P, P, S, nullptr, 1.0f);
        float* t = P; P = S; S = t;
    }
    // P now holds heat = expm(-T*adj), resident in workspace (not d_out).

    // Top-16 per row
    k_topk16<<<dim3(NMAT), blk, 0, stream>>>(P, idx);

    // Row mask bitset (2 MB), then symmetric mask via bit(i,j)|bit(j,i)
    k_zero_bits<<<dim3((unsigned)((NMAT * (NMAT / 32) + 255) / 256)), blk, 0, stream>>>(bits);
    k_scatter_bits<<<dim3((NMAT * TOPK) / 256), blk, 0, stream>>>(idx, bits);

    // Mask + symmetrize + I + row-normalize -> d_out
    k_finalize<<<dim3(NMAT), blk, 0, stream>>>(P, bits, out);
}